// NewsEntityGNN_678604832875
// MI455X (gfx1250) — compile-verified
//
#include <hip/hip_runtime.h>
#include <hip/hip_bf16.h>

typedef __attribute__((ext_vector_type(2))) float v2f;
typedef __attribute__((ext_vector_type(8))) float v8f;

// ---------------- utility kernels ----------------

__global__ void k_zero(float* __restrict__ p, int n) {
  int i = blockIdx.x * blockDim.x + threadIdx.x;
  if (i < n) p[i] = 0.0f;
}

__global__ void k_count(const int* __restrict__ dst, float* __restrict__ cnt, int ne) {
  int e = blockIdx.x * blockDim.x + threadIdx.x;
  if (e < ne) atomicAdd(&cnt[dst[e]], 1.0f);
}

__global__ void k_invcnt(float* __restrict__ cnt, int n) {
  int i = blockIdx.x * blockDim.x + threadIdx.x;
  if (i < n) cnt[i] = 1.0f / fmaxf(cnt[i], 1.0f);
}

// one block per edge, threads over feature dim -> coalesced gather + atomic scatter
__global__ void k_scatter(const float* __restrict__ x, const int* __restrict__ src,
                          const int* __restrict__ dst, float* __restrict__ msg,
                          int ne, int dim) {
  int e = blockIdx.x;
  int d = threadIdx.x;
  if (e < ne && d < dim) {
    int s = src[e];
    int t = dst[e];
    atomicAdd(&msg[(long long)t * dim + d], x[(long long)s * dim + d]);
  }
}

// ---------------- fused SAGE GEMM: out = (msg*rscale) @ Wl + Ar @ Wr + bias ----------------
// One wave (32 lanes) computes a 16x16 fp32 output tile via V_WMMA_F32_16X16X4_F32.
// A fragment (16x4 f32): lane l holds A[m0+(l&15)][k+2*(l>>4)+{0,1}] in {a.x,a.y}.
// B fragment (4x16 f32): lane l holds B[k+2*(l>>4)+{0,1}][n0+(l&15)].
// C/D (16x16 f32, v8f): element v of lane l is row m0+v+8*(l>>4), col n0+(l&15).
__global__ void k_sage_gemm(const float* __restrict__ Am, const float* __restrict__ rscale,
                            const float* __restrict__ Wl,
                            const float* __restrict__ Ar, const float* __restrict__ Wr,
                            const float* __restrict__ bias,
                            float* __restrict__ out, int M, int K, int N) {
  const int lane = threadIdx.x;           // 0..31, one wave per threadIdx.y row
  const int r    = lane >> 4;             // 0 or 1 (K-half / M-half selector)
  const int ml   = lane & 15;

  const int m0    = blockIdx.x * 16;
  const int ntile = blockIdx.y * blockDim.y + threadIdx.y;
  if (ntile * 16 >= N) return;            // whole-wave uniform exit
  const int n = ntile * 16 + ml;

  const int   mrow  = m0 + ml;
  const int   marow = (mrow < M) ? mrow : (M - 1);   // clamped, branchless
  const float mval  = (mrow < M) ? 1.0f : 0.0f;      // zero-fill padded rows
  const float sc    = rscale[marow] * mval;          // 1/deg for aggregated branch

  const float* __restrict__ ap = Am + (long long)marow * K;
  const float* __restrict__ xp = Ar + (long long)marow * K;

  v8f c = {};
  for (int kk = 0; kk < K; kk += 4) {
    const int ka = kk + 2 * r;
    v2f a, b;
    // aggregated-neighbor branch: (msg/deg) @ Wl
    a.x = ap[ka] * sc;
    a.y = ap[ka + 1] * sc;
    b.x = Wl[(long long)ka * N + n];
    b.y = Wl[(long long)(ka + 1) * N + n];
    c = __builtin_amdgcn_wmma_f32_16x16x4_f32(false, a, false, b, (short)0, c, false, false);
    // root branch: x @ Wr
    a.x = xp[ka] * mval;
    a.y = xp[ka + 1] * mval;
    b.x = Wr[(long long)ka * N + n];
    b.y = Wr[(long long)(ka + 1) * N + n];
    c = __builtin_amdgcn_wmma_f32_16x16x4_f32(false, a, false, b, (short)0, c, false, false);
  }

  const float bn = bias[n];
#pragma unroll
  for (int v = 0; v < 8; ++v) {
    const int row = m0 + v + 8 * r;
    if (row < M) out[(long long)row * N + n] = c[v] + bn;
  }
}

// ---------------- BatchNorm (training-mode, biased var) ----------------

__global__ void k_bn_stats(const float* __restrict__ h, float* __restrict__ stats,
                           int M, int rpb, int N) {
  int col = threadIdx.x;                  // one column per thread, N == blockDim.x
  int r0 = blockIdx.x * rpb;
  int r1 = r0 + rpb; if (r1 > M) r1 = M;
  float s = 0.0f, s2 = 0.0f;
  for (int row = r0; row < r1; ++row) {
    float v = h[(long long)row * N + col];
    s  += v;
    s2 += v * v;
  }
  atomicAdd(&stats[col], s);
  atomicAdd(&stats[N + col], s2);
}

__global__ void k_bn_apply(float* __restrict__ h, const float* __restrict__ stats,
                           const float* __restrict__ gamma, const float* __restrict__ beta,
                           int M, int N, int relu) {
  long long i = (long long)blockIdx.x * blockDim.x + threadIdx.x;
  if (i >= (long long)M * N) return;
  int col = (int)(i % N);
  float mean = stats[col] / (float)M;
  float var  = stats[N + col] / (float)M - mean * mean;
  float inv  = rsqrtf(var + 1e-5f);
  float v = (h[i] - mean) * inv * gamma[col] + beta[col];
  h[i] = relu ? fmaxf(v, 0.0f) : v;
}

// ---------------- launcher ----------------

extern "C" void kernel_launch(void* const* d_in, const int* in_sizes, int n_in,
                              void* d_out, int out_size, void* d_ws, size_t ws_size,
                              hipStream_t stream) {
  const float* x   = (const float*)d_in[0];
  const int*   ei  = (const int*)  d_in[1];
  const float* Wl0 = (const float*)d_in[2];
  const float* bl0 = (const float*)d_in[3];
  const float* Wr0 = (const float*)d_in[4];
  const float* g0  = (const float*)d_in[5];
  const float* be0 = (const float*)d_in[6];
  const float* Wl1 = (const float*)d_in[7];
  const float* bl1 = (const float*)d_in[8];
  const float* Wr1 = (const float*)d_in[9];
  const float* g1  = (const float*)d_in[10];
  const float* be1 = (const float*)d_in[11];
  float* out = (float*)d_out;

  const int HID  = in_sizes[3];           // 128
  const int IND  = in_sizes[2] / HID;     // 100
  const int NN   = in_sizes[0] / IND;     // 50000
  const int NE   = in_sizes[1] / 2;       // 800000
  const int OUTD = in_sizes[8];           // 128
  const int* src = ei;
  const int* dst = ei + NE;

  // workspace carve-up (fp32): msg[NN*HID] | cnt[NN] | h0[NN*HID] | stats[2*HID]
  float* msg   = (float*)d_ws;
  float* cnt   = msg + (size_t)NN * HID;
  float* h0    = cnt + NN;
  float* stats = h0 + (size_t)NN * HID;

  const int Z = 256;

  // ---- layer 0: SAGEConv -> BN -> ReLU ----
  {
    int n0 = NN * IND;
    k_zero<<<(n0 + Z - 1) / Z, Z, 0, stream>>>(msg, n0);
    k_zero<<<(NN + Z - 1) / Z, Z, 0, stream>>>(cnt, NN);
    k_zero<<<1, Z, 0, stream>>>(stats, 2 * HID);
  }
  k_count<<<(NE + Z - 1) / Z, Z, 0, stream>>>(dst, cnt, NE);
  k_scatter<<<NE, 128, 0, stream>>>(x, src, dst, msg, NE, IND);
  k_invcnt<<<(NN + Z - 1) / Z, Z, 0, stream>>>(cnt, NN);

  dim3 gblk(32, 4);                       // 4 waves/block, one 16x16 tile each
  dim3 ggrid0((NN + 15) / 16, (HID / 16 + 3) / 4);
  k_sage_gemm<<<ggrid0, gblk, 0, stream>>>(msg, cnt, Wl0, x, Wr0, bl0, h0, NN, IND, HID);

  const int rpb  = 128;
  const int nblk = (NN + rpb - 1) / rpb;
  k_bn_stats<<<nblk, HID, 0, stream>>>(h0, stats, NN, rpb, HID);
  {
    long long tot = (long long)NN * HID;
    k_bn_apply<<<(unsigned)((tot + Z - 1) / Z), Z, 0, stream>>>(h0, stats, g0, be0, NN, HID, 1);
  }

  // ---- layer 1: SAGEConv -> BN ----
  {
    int n1 = NN * HID;
    k_zero<<<(n1 + Z - 1) / Z, Z, 0, stream>>>(msg, n1);
    k_zero<<<1, Z, 0, stream>>>(stats, 2 * OUTD);
  }
  k_scatter<<<NE, 128, 0, stream>>>(h0, src, dst, msg, NE, HID);

  dim3 ggrid1((NN + 15) / 16, (OUTD / 16 + 3) / 4);
  k_sage_gemm<<<ggrid1, gblk, 0, stream>>>(msg, cnt, Wl1, h0, Wr1, bl1, out, NN, HID, OUTD);

  k_bn_stats<<<nblk, OUTD, 0, stream>>>(out, stats, NN, rpb, OUTD);
  {
    long long tot = (long long)NN * OUTD;
    k_bn_apply<<<(unsigned)((tot + Z - 1) / Z), Z, 0, stream>>>(out, stats, g1, be1, NN, OUTD, 0);
  }
}